// IMPO_31078383354673
// MI455X (gfx1250) — compile-verified
//
#include <hip/hip_runtime.h>

typedef float v2f __attribute__((ext_vector_type(2)));
typedef float v8f __attribute__((ext_vector_type(8)));

#define B_   32
#define N_   8192
#define F_   256
#define H_   4
#define K_   3
#define E_   262144
#define HK_  12
#define NC1_ 2

// ---------------------------------------------------------------------------
// Kernel A: Student-t soft assignment via fp32 WMMA (16x16x4).
// One wave handles a 16-point tile vs 16 centroids (12 real, 4 zero-padded).
// Writes dist [B,H,N,K] output, mask output, and S [B,N,K] workspace.
// ---------------------------------------------------------------------------
__global__ __launch_bounds__(256) void dist_assign_kernel(
    const float* __restrict__ x, const float* __restrict__ kmat,
    const unsigned char* __restrict__ mask,
    float* __restrict__ out_dist, float* __restrict__ out_mask,
    float* __restrict__ S_ws)
{
  __shared__ float kbuf[16 * F_];       // centroids, zero-padded rows 12..15
  __shared__ float knorm[16];
  __shared__ float distS[8][16][16];    // per-wave dist tiles

  const int t = threadIdx.x;
  for (int i = t; i < 16 * F_; i += 256)
    kbuf[i] = (i < HK_ * F_) ? kmat[i] : 0.0f;
  __syncthreads();
  if (t < 16) {
    float s = 0.0f;
    for (int c = 0; c < F_; ++c) { float v = kbuf[t * F_ + c]; s += v * v; }
    knorm[t] = s;
  }
  __syncthreads();

  const int wave  = t >> 5;
  const int lane  = t & 31;
  const int tile  = blockIdx.x * 8 + wave;   // global 16-point tile id
  const int p0    = tile * 16;
  const int b     = p0 >> 13;                // / 8192
  const int n0    = p0 & (N_ - 1);

  const int mrow  = lane & 15;               // A: point row / B: centroid col
  const int khalf = (lane >> 4) << 1;        // K offset 0 or 2 per half-wave
  const float* xrow = x + ((size_t)b * N_ + (size_t)(n0 + mrow)) * F_;

  v8f acc = {};
  float axx = 0.0f;                          // partial row norm of x
  for (int kk = 0; kk < F_; kk += 4) {
    const int koff = kk + khalf;
    float2 av = *(const float2*)(xrow + koff);
    float2 bv = *(const float2*)(&kbuf[mrow * F_ + koff]);
    v2f a;  a.x = av.x;  a.y = av.y;
    v2f bm; bm.x = bv.x; bm.y = bv.y;
    axx += av.x * av.x + av.y * av.y;
    acc = __builtin_amdgcn_wmma_f32_16x16x4_f32(false, a, false, bm,
                                                (short)0, acc, false, false);
  }

  // complete |x_m|^2 across the two half-waves; lane L holds xnorm[L&15]
  float xn_row = axx + __shfl_xor(axx, 16, 32);
  const float kn = knorm[lane & 15];
#pragma unroll
  for (int r = 0; r < 8; ++r) {
    const int m = r + ((lane >> 4) << 3);
    float xn = __shfl(xn_row, m, 32);
    float d2 = kn + xn - 2.0f * acc[r];
    d2 = fmaxf(d2, 0.0f);
    distS[wave][m][lane & 15] = 1.0f / (1.0f + d2);   // (1+d2)^-1, TAU=1
  }
  __syncthreads();

  // per-head normalize over K=3, sum heads -> S; apply mask
  const int pblk = blockIdx.x * 128;
  const int bb   = pblk >> 13;
  const int nb0  = pblk & (N_ - 1);
  if (t < 128) {
    const int w = t >> 4, m = t & 15;
    const int n = nb0 + t;
    const float mb = mask[(size_t)bb * N_ + n] ? 1.0f : 0.0f;
    float s0 = 0.0f, s1 = 0.0f, s2 = 0.0f;
#pragma unroll
    for (int h = 0; h < H_; ++h) {
      float d0 = distS[w][m][h * 3 + 0];
      float d1 = distS[w][m][h * 3 + 1];
      float d2v = distS[w][m][h * 3 + 2];
      float inv = 1.0f / (d0 + d1 + d2v);
      s0 += d0 * inv; s1 += d1 * inv; s2 += d2v * inv;
    }
    const size_t base = ((size_t)bb * N_ + n) * 3;
    S_ws[base + 0] = s0 * mb;
    S_ws[base + 1] = s1 * mb;
    S_ws[base + 2] = s2 * mb;
    out_mask[(size_t)bb * N_ + n] = mb;
  }
  // dist output [B,H,N,K]
  for (int idx = t; idx < 128 * 12; idx += 256) {
    const int p = idx / 12, c = idx % 12;
    const int h = c / 3, k2 = c % 3;
    const int n = nb0 + p;
    out_dist[(((size_t)bb * H_ + h) * N_ + n) * 3 + k2] = distS[p >> 4][p & 15][c];
  }
}

// ---------------------------------------------------------------------------
__global__ __launch_bounds__(256) void zero_kernel(float* __restrict__ p, int n)
{
  int i = blockIdx.x * 256 + threadIdx.x;
  if (i < n) p[i] = 0.0f;
}

// Kernel C: per-graph sparse aggregation via global atomics.
__global__ __launch_bounds__(256) void edge_agg_kernel(
    const int* __restrict__ ei, const float* __restrict__ S_ws,
    float* __restrict__ S_agg, float* __restrict__ deg)
{
  size_t gid = (size_t)blockIdx.x * 256 + threadIdx.x;   // b*E + e
  if (gid >= (size_t)B_ * E_) return;
  const int b = (int)(gid >> 18);                        // E_ = 2^18
  const int e = (int)(gid & (E_ - 1));
  const int* eb = ei + (size_t)b * 2 * E_;
  const int src = eb[e];
  const int dst = eb[E_ + e];
  const float* sd = S_ws + ((size_t)b * N_ + dst) * 3;
  float* so = S_agg + ((size_t)b * N_ + src) * 3;
  atomicAdd(&so[0], sd[0]);
  atomicAdd(&so[1], sd[1]);
  atomicAdd(&so[2], sd[2]);
  atomicAdd(&deg[(size_t)b * N_ + src], 1.0f);
  atomicAdd(&deg[(size_t)b * N_ + dst], 1.0f);
}

// Kernel D: degree-normalize, softmax over K=3, hard one-hot mask.
__global__ __launch_bounds__(256) void softmax_hard_kernel(
    const float* __restrict__ S_agg, const float* __restrict__ deg,
    float* __restrict__ out_Shard, float* __restrict__ out_Sraw,
    float* __restrict__ S4_ws)
{
  size_t i = (size_t)blockIdx.x * 256 + threadIdx.x;     // b*N + n
  if (i >= (size_t)B_ * N_) return;
  float d = 0.5f * deg[i];
  if (d == 0.0f) d = 1.0f;
  const float inv = 1.0f / d;
  const float r0 = S_agg[i * 3 + 0] * inv;
  const float r1 = S_agg[i * 3 + 1] * inv;
  const float r2 = S_agg[i * 3 + 2] * inv;
  out_Sraw[i * 3 + 0] = r0;
  out_Sraw[i * 3 + 1] = r1;
  out_Sraw[i * 3 + 2] = r2;
  const float mx = fmaxf(r0, fmaxf(r1, r2));
  const float e0 = expf(r0 - mx), e1 = expf(r1 - mx), e2 = expf(r2 - mx);
  const float isum = 1.0f / (e0 + e1 + e2);
  const float g0 = e0 * isum, g1 = e1 * isum, g2 = e2 * isum;
  int am = 0; float best = g0;
  if (g1 > best) { best = g1; am = 1; }
  if (g2 > best) { best = g2; am = 2; }
  const float h0 = (am == 0) ? g0 : 0.0f;
  const float h1 = (am == 1) ? g1 : 0.0f;
  const float h2 = (am == 2) ? g2 : 0.0f;
  out_Shard[i * 3 + 0] = h0;
  out_Shard[i * 3 + 1] = h1;
  out_Shard[i * 3 + 2] = h2;
  S4_ws[i * 4 + 0] = h0;
  S4_ws[i * 4 + 1] = h1;
  S4_ws[i * 4 + 2] = h2;
  S4_ws[i * 4 + 3] = 0.0f;                               // zero pad cluster 3
}

// ---------------------------------------------------------------------------
// Kernel E: pooled^T[f][c] = sum_n x[n][f] * S_hard[n][c] via fp32 WMMA.
// One wave per (batch, 16-feature tile); K-loop over n in steps of 4.
// ---------------------------------------------------------------------------
__global__ __launch_bounds__(256) void pooled_wmma_kernel(
    const float* __restrict__ x, const float* __restrict__ S4,
    float* __restrict__ pooledT)
{
  const int wid   = blockIdx.x * 8 + (threadIdx.x >> 5); // 0..511
  const int lane  = threadIdx.x & 31;
  const int b     = wid >> 4;
  const int ftile = wid & 15;
  const int fi    = ftile * 16 + (lane & 15);            // A: feature row
  const int c     = lane & 15;                           // B: cluster col
  const int koff  = (lane >> 4) << 1;                    // 0 or 2
  const float* xb  = x  + (size_t)b * N_ * F_;
  const float* s4b = S4 + (size_t)b * N_ * 4;

  v8f acc = {};
  for (int n0 = 0; n0 < N_; n0 += 4) {
    v2f a, bm;
    a.x  = xb[(size_t)(n0 + koff)     * F_ + fi];
    a.y  = xb[(size_t)(n0 + koff + 1) * F_ + fi];
    bm.x = (c < 4) ? s4b[(size_t)(n0 + koff)     * 4 + c] : 0.0f;
    bm.y = (c < 4) ? s4b[(size_t)(n0 + koff + 1) * 4 + c] : 0.0f;
    acc = __builtin_amdgcn_wmma_f32_16x16x4_f32(false, a, false, bm,
                                                (short)0, acc, false, false);
  }
#pragma unroll
  for (int r = 0; r < 8; ++r) {
    const int f = ftile * 16 + r + ((lane >> 4) << 3);
    pooledT[((size_t)b * 16 + c) * F_ + f] = acc[r];
  }
}

// Kernel F: xp[b,c,fo] = sum_fi pooled[b,c,fi] * W[fo,fi], c < NC1.
__global__ __launch_bounds__(256) void final_kernel(
    const float* __restrict__ pooledT, const float* __restrict__ W,
    float* __restrict__ out_xp)
{
  const int b  = blockIdx.x;
  const int fo = threadIdx.x;
  const float* wrow = W + (size_t)fo * F_;
#pragma unroll
  for (int c = 0; c < NC1_; ++c) {
    const float* p = pooledT + ((size_t)b * 16 + c) * F_;
    float s = 0.0f;
    for (int f = 0; f < F_; ++f) s += p[f] * wrow[f];
    out_xp[((size_t)b * NC1_ + c) * F_ + fo] = s;
  }
}

// ---------------------------------------------------------------------------
extern "C" void kernel_launch(void* const* d_in, const int* in_sizes, int n_in,
                              void* d_out, int out_size, void* d_ws, size_t ws_size,
                              hipStream_t stream)
{
  const float*         x    = (const float*)d_in[0];
  const int*           ei   = (const int*)d_in[1];
  const unsigned char* mask = (const unsigned char*)d_in[2];
  const float*         km   = (const float*)d_in[3];
  const float*         W    = (const float*)d_in[4];
  float* out = (float*)d_out;

  // workspace layout (floats)
  float* S_ws    = (float*)d_ws;           // B*N*3 = 786432
  float* S_agg   = S_ws   + 786432;        // 786432
  float* deg     = S_agg  + 786432;        // 262144
  float* S4      = deg    + 262144;        // B*N*4 = 1048576
  float* pooledT = S4     + 1048576;       // B*16*F = 131072

  // output layout (return-order concat)
  float* out_xp    = out;                  // 32*2*256   = 16384
  float* out_Shard = out + 16384;          // 786432
  float* out_Sraw  = out_Shard + 786432;   // 786432
  float* out_dist  = out_Sraw  + 786432;   // 32*4*8192*3 = 3145728
  float* out_mask  = out_dist  + 3145728;  // 262144

  zero_kernel<<<(786432 + 262144 + 255) / 256, 256, 0, stream>>>(
      S_agg, 786432 + 262144);
  dist_assign_kernel<<<2048, 256, 0, stream>>>(
      x, km, mask, out_dist, out_mask, S_ws);
  edge_agg_kernel<<<(int)(((size_t)B_ * E_ + 255) / 256), 256, 0, stream>>>(
      ei, S_ws, S_agg, deg);
  softmax_hard_kernel<<<(B_ * N_ + 255) / 256, 256, 0, stream>>>(
      S_agg, deg, out_Shard, out_Sraw, S4);
  pooled_wmma_kernel<<<64, 256, 0, stream>>>(x, S4, pooledT);
  final_kernel<<<B_, 256, 0, stream>>>(pooledT, W, out_xp);
}